// MiniBatchDiscrimination_11759620457095
// MI455X (gfx1250) — compile-verified
//
#include <hip/hip_runtime.h>

typedef float v2f __attribute__((ext_vector_type(2)));
typedef float v8f __attribute__((ext_vector_type(8)));

#define Nn 256
#define Aa 2048
#define Bb 64
#define Cc 32
#define BC 2048   // Bb*Cc = GEMM N dimension
#define LPAD 33   // padded LDS row stride (floats) -> conflict-free on 64 banks
#define LROW (Bb * LPAD)  // 2112 floats per staged j-row

// ---------------------------------------------------------------------------
// Kernel 1: Ms[256, 2048] = x[256, 2048] @ T[2048, 2048]
// f32 WMMA (V_WMMA_F32_16X16X4_F32), 32x32 output macro-tile per wave:
// 2 A-fragments x 2 B-fragments -> 4 independent WMMAs per K-step.
// Halves L2 traffic vs one-tile-per-wave and doubles FLOP per load.
// ---------------------------------------------------------------------------
__global__ __launch_bounds__(256) void mbd_gemm_f32wmma(
    const float* __restrict__ x, const float* __restrict__ T,
    float* __restrict__ Ms) {
  const int wave = threadIdx.x >> 5;
  const int lane = threadIdx.x & 31;
  const int half = lane >> 4;  // selects K pair {k,k+1} vs {k+2,k+3}
  const int r    = lane & 15;

  const int mt = blockIdx.x >> 3;              // 0..7  -> 32-row M tile
  const int ng = blockIdx.x & 7;               // 0..7  -> group of 8 N tiles
  const int m0 = mt * 32;
  const int n0 = (ng * 8 + wave) * 32;

  const float* xrow0 = x + (size_t)(m0 + r) * Aa;        // A rows m0   + r
  const float* xrow1 = x + (size_t)(m0 + 16 + r) * Aa;   // A rows m0+16+ r
  const float* tcol0 = T + n0 + r;                       // B cols n0   + r
  const float* tcol1 = T + n0 + 16 + r;                  // B cols n0+16+ r

  v8f c00 = {}, c01 = {}, c10 = {}, c11 = {};
#pragma unroll 2
  for (int k = 0; k < Aa; k += 4) {
    const int ka = k + 2 * half;
    v2f a0, a1, b0, b1;
    a0.x = xrow0[ka];             a0.y = xrow0[ka + 1];
    a1.x = xrow1[ka];             a1.y = xrow1[ka + 1];
    b0.x = tcol0[(size_t)ka * BC]; b0.y = tcol0[(size_t)(ka + 1) * BC];
    b1.x = tcol1[(size_t)ka * BC]; b1.y = tcol1[(size_t)(ka + 1) * BC];
    c00 = __builtin_amdgcn_wmma_f32_16x16x4_f32(false, a0, false, b0,
                                                (short)0, c00, false, false);
    c01 = __builtin_amdgcn_wmma_f32_16x16x4_f32(false, a0, false, b1,
                                                (short)0, c01, false, false);
    c10 = __builtin_amdgcn_wmma_f32_16x16x4_f32(false, a1, false, b0,
                                                (short)0, c10, false, false);
    c11 = __builtin_amdgcn_wmma_f32_16x16x4_f32(false, a1, false, b1,
                                                (short)0, c11, false, false);
  }

  // C/D layout: VGPR g, lanes 0-15 -> M=g, lanes 16-31 -> M=g+8; N = lane%16
  {
    float* o00 = Ms + (size_t)(m0 + 8 * half) * BC + n0 + r;
    float* o10 = Ms + (size_t)(m0 + 16 + 8 * half) * BC + n0 + r;
#pragma unroll
    for (int g = 0; g < 8; ++g) {
      o00[(size_t)g * BC]      = c00[g];
      o00[(size_t)g * BC + 16] = c01[g];
      o10[(size_t)g * BC]      = c10[g];
      o10[(size_t)g * BC + 16] = c11[g];
    }
  }
}

// ---------------------------------------------------------------------------
// Kernel 2: out[i,b] = sum_j exp(-sum_c |Ms[i,b,c] - Ms[j,b,c]|)
// Block handles 4 consecutive i rows (Ms[i,b,:] in registers).
// 4 j-rows staged per LDS round (stride-33 padded) -> 4x fewer barriers,
// batched b128 global loads.
// ---------------------------------------------------------------------------
__global__ __launch_bounds__(256) void mbd_l1exp(
    const float* __restrict__ Ms, float* __restrict__ out) {
  __shared__ float lds[4 * LROW];  // 4 staged j-rows, 33.8 KB

  const int t  = threadIdx.x;
  const int il = t >> 6;       // 0..3 local i
  const int b  = t & 63;       // feature b
  const int i  = blockIdx.x * 4 + il;

  // My row's slice Ms[i, b, 0..31] in registers (exact f32)
  float mi[Cc];
  const float* mrow = Ms + (size_t)i * BC + b * Cc;
#pragma unroll
  for (int c4 = 0; c4 < Cc; c4 += 4) {
    float4 v = *(const float4*)(mrow + c4);
    mi[c4 + 0] = v.x; mi[c4 + 1] = v.y; mi[c4 + 2] = v.z; mi[c4 + 3] = v.w;
  }

  // Each thread's fixed slot in the staging pattern: 8 contiguous floats
  const int e  = t * 8;        // multiple of 8
  const int bb = e >> 5;       // destination b-row
  const int cc = e & 31;       // 0/8/16/24 -> never crosses padded row
  const int dstoff = bb * LPAD + cc;

  float acc = 0.0f;
  for (int jt = 0; jt < Nn / 4; ++jt) {
    __syncthreads();
#pragma unroll
    for (int rr = 0; rr < 4; ++rr) {
      const float* src = Ms + (size_t)(4 * jt + rr) * BC + e;
      float4 v0 = *(const float4*)(src);
      float4 v1 = *(const float4*)(src + 4);
      float* dst = lds + rr * LROW + dstoff;
      dst[0] = v0.x; dst[1] = v0.y; dst[2] = v0.z; dst[3] = v0.w;
      dst[4] = v1.x; dst[5] = v1.y; dst[6] = v1.z; dst[7] = v1.w;
    }
    __syncthreads();

#pragma unroll
    for (int rr = 0; rr < 4; ++rr) {
      const float* mj = lds + rr * LROW + b * LPAD;
      float l1 = 0.0f;
#pragma unroll
      for (int c = 0; c < Cc; ++c) l1 += fabsf(mi[c] - mj[c]);
      acc += expf(-l1);
    }
  }

  out[(size_t)i * Bb + b] = acc;
}

// ---------------------------------------------------------------------------
extern "C" void kernel_launch(void* const* d_in, const int* in_sizes, int n_in,
                              void* d_out, int out_size, void* d_ws,
                              size_t ws_size, hipStream_t stream) {
  const float* x = (const float*)d_in[0];   // [256, 2048] f32
  const float* T = (const float*)d_in[1];   // [2048, 64, 32] f32 = [2048, 2048]
  float* out = (float*)d_out;               // [256, 64] f32
  float* Ms  = (float*)d_ws;                // [256, 2048] f32 scratch (2 MB)

  // GEMM: 8 M-tiles(32) x 8 wave-groups (8 waves x 32-col N tiles each)
  mbd_gemm_f32wmma<<<64, 256, 0, stream>>>(x, T, Ms);
  // Pairwise L1 + exp reduction: 64 blocks x 4 i-rows
  mbd_l1exp<<<64, 256, 0, stream>>>(Ms, out);
}